// CRFLoss_24352464569066
// MI455X (gfx1250) — compile-verified
//
#include <hip/hip_runtime.h>
#include <hip/hip_bf16.h>
#include <stdint.h>

// ---------------------------------------------------------------------------
// CRF forward-algorithm loss for MI455X (gfx1250).
//   scores: (L=256, B=128, T=64, T=64) f32   -> streamed once, 512 MB
//   target: (L, B) int                       -> gather
//   mask  : (L, B) bool                      -> all-ones in setup
// One workgroup per batch element b (128 WGs, 8 wave32 each).
// Per step l: TDM-prefetch tile l+1 into LDS while computing on tile l.
// logsumexp step done as: E' = exp(score + part[i] - M), colsum via
// V_WMMA_F32_16X16X4_F32 with an all-ones A matrix, then part = M + log(colsum).
// ---------------------------------------------------------------------------

typedef __attribute__((ext_vector_type(2))) float v2f;
typedef __attribute__((ext_vector_type(8))) float v8f;
typedef __attribute__((ext_vector_type(4))) unsigned int u32x4;
typedef __attribute__((ext_vector_type(8))) int i32x8;
typedef __attribute__((ext_vector_type(4))) int i32x4;

#define LL 256
#define BB 128
#define TT 64
#define START_TAG 62
#define END_TAG 63

// Issue a TDM DMA of one contiguous 64x64 f32 tile (16 KB) global -> LDS.
// Descriptor layout per CDNA5 ISA ch.8 (D# groups).
__device__ __forceinline__ void tdm_load_tile(const float* gbase,
                                              unsigned long long elem_off,
                                              unsigned int lds_byte) {
  unsigned long long ga = (unsigned long long)(uintptr_t)gbase + elem_off * 4ull;
  u32x4 g0;
  g0[0] = 1u;                                   // count=1 (valid descriptor)
  g0[1] = lds_byte;                             // lds_addr
  g0[2] = (unsigned int)ga;                     // global_addr[31:0]
  g0[3] = (unsigned int)((ga >> 32) & 0x1FFFFFFull) | 0x80000000u; // [56:32] + type=2
  i32x8 g1;
  g1[0] = (int)(2u << 16);        // data_size = 2 (4 bytes); wg_mask = 0
  g1[1] = (int)(64u << 16);       // tensor_dim0[15:0] = 64 (atomic_barrier_addr=0)
  g1[2] = (int)(64u << 16);       // tensor_dim0 hi=0, tensor_dim1[15:0] = 64
  g1[3] = (int)(64u << 16);       // tensor_dim1 hi=0, tile_dim0 = 64
  g1[4] = (int)64;                // tile_dim1 = 64, tile_dim2 = 0
  g1[5] = (int)64;                // tensor_dim0_stride[31:0] = 64 elements
  g1[6] = 0;                      // stride0 hi, stride1 lo = 0
  g1[7] = 0;
  i32x4 gz4 = {0, 0, 0, 0};
  i32x8 gz8 = {0, 0, 0, 0, 0, 0, 0, 0};
  __builtin_amdgcn_tensor_load_to_lds(g0, g1, gz4, gz4, gz8, 0);
}

__global__ void __launch_bounds__(256)
crf_forward_kernel(const float* __restrict__ scores,
                   const int* __restrict__ target,
                   const unsigned char* __restrict__ mask,
                   float* __restrict__ out) {
  __shared__ float tile[2][TT * TT];   // 2 x 16 KB double buffer
  __shared__ float part[TT];
  __shared__ float colsum[TT];
  __shared__ float tg_shared;

  const int b    = blockIdx.x;
  const int t    = threadIdx.x;
  const int lane = t & 31;
  const int wave = t >> 5;

  if (t == 0) tg_shared = 0.0f;
  float tg_acc = 0.0f;

  // Prefetch tiles for l=0 and l=1; wait for tile 0 (in-order completion).
  if (wave == 0) {
    tdm_load_tile(scores, ((unsigned long long)0 * BB + b) * (TT * TT),
                  (unsigned int)(uintptr_t)&tile[0][0]);
    tdm_load_tile(scores, ((unsigned long long)1 * BB + b) * (TT * TT),
                  (unsigned int)(uintptr_t)&tile[1][0]);
    __builtin_amdgcn_s_wait_tensorcnt(1);
  }
  __syncthreads();

  // l = 0: init partition from START_TAG row; gather target energy.
  if (t < TT) part[t] = tile[0][START_TAG * TT + t];
  if (t == 0) {
    const int tgt0 = target[0 * BB + b];
    tg_acc += mask[0 * BB + b] ? tile[0][tgt0] : 0.0f;
  }
  __syncthreads();

  for (int l = 1; l < LL; ++l) {
    const int buf = l & 1;

    // Everyone is past the previous iteration (barrier above), so buf^1 may be
    // overwritten: issue prefetch for l+1, then wait for tile l.
    if (wave == 0) {
      if (l + 1 < LL) {
        tdm_load_tile(scores, ((unsigned long long)(l + 1) * BB + b) * (TT * TT),
                      (unsigned int)(uintptr_t)&tile[buf ^ 1][0]);
        __builtin_amdgcn_s_wait_tensorcnt(1);
      } else {
        __builtin_amdgcn_s_wait_tensorcnt(0);
      }
    }
    __syncthreads();  // tile l visible to all waves

    // Stabilizer M = max_i part[i]  (64 broadcast LDS reads, redundant per thread)
    float M = -3.402823466e+38f;
#pragma unroll
    for (int i = 0; i < TT; ++i) M = fmaxf(M, part[i]);

    const int tgt          = target[l * BB + b];
    const unsigned char ml = mask[l * BB + b];

    // E' phase: tile[idx] = exp(score + part[i] - M); fold in the gather
    // (owner thread reads its raw element before overwriting it).
#pragma unroll
    for (int e = 0; e < 16; ++e) {
      const int idx = e * 256 + t;           // 4096 elements / 256 threads
      const float raw = tile[buf][idx];
      if (idx == tgt) tg_acc += ml ? raw : 0.0f;
      const int i = idx >> 6;                // row (uniform per wave)
      tile[buf][idx] = __expf(raw + part[i] - M);
    }
    __syncthreads();  // E' ready

    // Column sums via WMMA: waves 0..3 each own a 16-wide j-tile.
    // D = A(ones,16x4) x B(E' 4x16) + C  => every row of D = column sums.
    if (wave < 4) {
      const int j0 = wave * 16;
      const int n  = lane & 15;
      const int hi = lane >> 4;              // lanes 16-31 carry the other K pair
      v8f c = {0.f, 0.f, 0.f, 0.f, 0.f, 0.f, 0.f, 0.f};
      v2f a;
      a[0] = 1.0f;
      a[1] = 1.0f;
#pragma unroll
      for (int ch = 0; ch < 16; ++ch) {
        const int k = ch * 4 + 2 * hi;
        v2f bm;
        bm[0] = tile[buf][(k + 0) * TT + j0 + n];
        bm[1] = tile[buf][(k + 1) * TT + j0 + n];
        c = __builtin_amdgcn_wmma_f32_16x16x4_f32(false, a, false, bm,
                                                  (short)0, c, false, false);
      }
      if (lane < 16) colsum[j0 + lane] = c[0];  // row M=0: lanes 0-15, VGPR0
    }
    __syncthreads();

    // partition update (masked)
    if (t < TT && ml) part[t] = M + __logf(colsum[t]);
    __syncthreads();
  }

  // Block reduction of the gather energy; one atomic per block into d_out.
  atomicAdd(&tg_shared, tg_acc);
  __syncthreads();
  if (t == 0) {
    atomicAdd(out, (part[END_TAG] - tg_shared) * (1.0f / (float)BB));
  }
}

__global__ void zero_out_kernel(float* out) { out[0] = 0.0f; }

extern "C" void kernel_launch(void* const* d_in, const int* in_sizes, int n_in,
                              void* d_out, int out_size, void* d_ws, size_t ws_size,
                              hipStream_t stream) {
  const float*         scores = (const float*)d_in[0];
  const int*           target = (const int*)d_in[1];
  const unsigned char* mask   = (const unsigned char*)d_in[2];
  float*               out    = (float*)d_out;
  (void)in_sizes; (void)n_in; (void)out_size; (void)d_ws; (void)ws_size;

  zero_out_kernel<<<1, 1, 0, stream>>>(out);
  crf_forward_kernel<<<BB, 256, 0, stream>>>(scores, target, mask, out);
}